// lstm_11055245820200
// MI455X (gfx1250) — compile-verified
//
#include <hip/hip_runtime.h>
#include <math.h>

#define T_SEQ 8192
#define IN_DIM 40
#define E_DIM 256
#define H_DIM 512
#define G_DIM 2048    // 4*H
#define NWG_ROLE 64   // WGs per role; 3 roles => 192 WGs
#define NWG_ALL 192
#define S_STEPS (T_SEQ + 2)   // pipeline steps (layer-1 cell lags by 2)

typedef float v2f __attribute__((ext_vector_type(2)));
typedef float v8f __attribute__((ext_vector_type(8)));

// ---------------------------------------------------------------------------
// prep: WcT[k][r] = (W_ih0 @ W_in)[r][k]  (K-major [40][2048])
//       bc0 = b_ih0+b_hh0 + W_ih0@b_in,  bc1 = b_ih1+b_hh1
// ---------------------------------------------------------------------------
__global__ void prep_kernel(const float* __restrict__ W_in, const float* __restrict__ b_in,
                            const float* __restrict__ W_ih0,
                            const float* __restrict__ b_ih0, const float* __restrict__ b_hh0,
                            const float* __restrict__ b_ih1, const float* __restrict__ b_hh1,
                            float* __restrict__ WcT, float* __restrict__ bc0,
                            float* __restrict__ bc1) {
    int idx = blockIdx.x * blockDim.x + threadIdx.x;
    const int nWc = G_DIM * IN_DIM;
    if (idx < nWc) {
        int r = idx / IN_DIM, k = idx % IN_DIM;
        float s = 0.f;
        for (int e = 0; e < E_DIM; ++e) s += W_ih0[r * E_DIM + e] * W_in[e * IN_DIM + k];
        WcT[(size_t)k * G_DIM + r] = s;                 // K-major
    } else if (idx < nWc + G_DIM) {
        int r = idx - nWc;
        float s = b_ih0[r] + b_hh0[r];
        for (int e = 0; e < E_DIM; ++e) s += W_ih0[r * E_DIM + e] * b_in[e];
        bc0[r] = s;
    } else if (idx < nWc + 2 * G_DIM) {
        int r = idx - nWc - G_DIM;
        bc1[r] = b_ih1[r] + b_hh1[r];
    }
}

// ---------------------------------------------------------------------------
// tiled transpose: src[R][C] -> dst[C][R]
// ---------------------------------------------------------------------------
__global__ __launch_bounds__(256)
void transpose_kernel(const float* __restrict__ src, float* __restrict__ dst,
                      int R, int C) {
    __shared__ float tile[32][33];
    const int c0 = blockIdx.x * 32, r0 = blockIdx.y * 32;
    const int tx = threadIdx.x & 31, ty = threadIdx.x >> 5;
    for (int i = ty; i < 32; i += 8) {
        int r = r0 + i, c = c0 + tx;
        tile[i][tx] = (r < R && c < C) ? src[(size_t)r * C + c] : 0.f;
    }
    __syncthreads();
    for (int i = ty; i < 32; i += 8) {
        int c = c0 + i, r = r0 + tx;
        if (c < C && r < R) dst[(size_t)c * R + r] = tile[tx][i];
    }
}

__global__ void reset_kernel(float* __restrict__ h1buf, float* __restrict__ h2buf,
                             int* __restrict__ flags) {
    int idx = blockIdx.x * blockDim.x + threadIdx.x;
    if (idx < 2 * H_DIM) { h1buf[idx] = 0.f; h2buf[idx] = 0.f; }
    if (idx < 256) flags[idx] = 0;
}

__global__ void out_init_kernel(float* __restrict__ outp, const float* __restrict__ b_out) {
    int idx = blockIdx.x * blockDim.x + threadIdx.x;
    if (idx < T_SEQ) outp[idx] = b_out[0];
}

// ---------------------------------------------------------------------------
// WMMA f32 GEMM with K-major operands: C[M,N] = AT^T @ WT + bias
// One wave -> 16x64 tile via V_WMMA_F32_16X16X4_F32; depth-1 SW pipeline.
// ---------------------------------------------------------------------------
__global__ __launch_bounds__(128)
void wmma_gemm_kmajor(const float* __restrict__ AT,   // [K][M]
                      const float* __restrict__ WT,   // [K][N]
                      const float* __restrict__ bias, // [N]
                      float* __restrict__ C,          // [M][N]
                      int M, int N, int K) {
    const int lane = threadIdx.x & 31;
    const int wib  = threadIdx.x >> 5;
    const int wavesPerRow = N / 64;
    const int wid = blockIdx.x * 4 + wib;
    const int mt  = wid / wavesPerRow;
    const int ng  = wid % wavesPerRow;
    if (mt * 16 >= M) return;                 // wave-uniform

    const int r  = lane & 15;
    const int kh = (lane >> 4) * 2;

    const float* Ap = AT + (size_t)kh * M + (mt * 16 + r);
    const float* Bp = WT + (size_t)kh * N + (ng * 64 + r);

    v8f acc[4];
    acc[0] = v8f{}; acc[1] = v8f{}; acc[2] = v8f{}; acc[3] = v8f{};

    const int steps = K >> 2;
    v2f a_c, b_c[4], a_n, b_n[4];
    a_c.x = Ap[0]; a_c.y = Ap[M];
#pragma unroll
    for (int g = 0; g < 4; ++g) { b_c[g].x = Bp[g * 16]; b_c[g].y = Bp[N + g * 16]; }
    a_n = a_c;
#pragma unroll
    for (int g = 0; g < 4; ++g) b_n[g] = b_c[g];

    for (int ks = 0; ks < steps; ++ks) {
        if (ks + 1 < steps) {
            const float* An = Ap + (size_t)(ks + 1) * 4 * M;
            const float* Bn = Bp + (size_t)(ks + 1) * 4 * N;
            a_n.x = An[0]; a_n.y = An[M];
#pragma unroll
            for (int g = 0; g < 4; ++g) { b_n[g].x = Bn[g * 16]; b_n[g].y = Bn[N + g * 16]; }
        }
#pragma unroll
        for (int g = 0; g < 4; ++g)
            acc[g] = __builtin_amdgcn_wmma_f32_16x16x4_f32(
                false, a_c, false, b_c[g], (short)0, acc[g], false, false);
        a_c = a_n;
#pragma unroll
        for (int g = 0; g < 4; ++g) b_c[g] = b_n[g];
    }

    const int mrow0 = mt * 16 + (lane >> 4) * 8;
#pragma unroll
    for (int g = 0; g < 4; ++g) {
        const int n = ng * 64 + g * 16 + r;
        const float bv = bias ? bias[n] : 0.f;
#pragma unroll
        for (int v = 0; v < 8; ++v)
            C[(size_t)(mrow0 + v) * N + n] = acc[g][v] + bv;
    }
}

// ---------------------------------------------------------------------------
// Fused persistent two-layer scan: 192 WGs x 256 threads, one barrier chain.
//   role 0 (WG   0- 63): layer-0 cell,  t = s      (reads xg0, writes h1)
//   role 1 (WG  64-127): ig = W_ih1@h1, t = s-1    (writes igbuf)
//   role 2 (WG 128-191): layer-1 cell,  t = s-2    (reads igbuf, writes h2, out)
// Each WG owns 8 h-lanes => 32 gate rows x 512 K in LDS, k-major [512][32].
// Parity scheme: h published at parity s&1, read at (s+1)&1;
//                ig written at parity (s+1)&1, read at s&1.
// Barrier: flags[wg] = s+1 (release), wave0 lanes poll 6 flags each.
// ---------------------------------------------------------------------------
__global__ __launch_bounds__(256)
void lstm_fused_scan(const float* __restrict__ xg0,   // [T, 2048]
                     const float* __restrict__ Whh0,  // [2048, 512]
                     const float* __restrict__ Wih1,  // [2048, 512]
                     const float* __restrict__ Whh1,  // [2048, 512]
                     const float* __restrict__ bc1,   // [2048]
                     const float* __restrict__ Wout,  // [512]
                     float* __restrict__ h1buf,       // [2,512] zeroed
                     float* __restrict__ h2buf,       // [2,512] zeroed
                     float* __restrict__ igbuf,       // [2,2048]
                     float* __restrict__ outp,        // [T] pre-set to b_out
                     float* __restrict__ hT0, float* __restrict__ cT0,
                     float* __restrict__ hT1, float* __restrict__ cT1,
                     int* __restrict__ flags) {
    __shared__ float wlds[H_DIM][32];   // 64KB, [k][local-row]
    __shared__ float hlds[H_DIM];
    __shared__ float pt[8][32];
    __shared__ float glds[32];
    __shared__ float hnew[8];

    const int tid  = threadIdx.x;
    const int wg   = blockIdx.x;
    const int role = wg >> 6;           // 0,1,2
    const int slc  = wg & 63;
    const int hb   = slc * 8;
    const int lrow = tid & 31;
    const int ks   = tid >> 5;

    const float* Wsrc = (role == 0) ? Whh0 : (role == 1) ? Wih1 : Whh1;
    {   // stage this WG's 32 gate rows, k-major
        const int grow = (lrow >> 3) * H_DIM + hb + (lrow & 7);
        const float* wr = Wsrc + (size_t)grow * H_DIM;
        for (int k = ks * 64; k < ks * 64 + 64; ++k) wlds[k][lrow] = wr[k];
    }
    __syncthreads();

    float breg = 0.f;                   // role-2 bias cache
    if (role == 2 && tid < 32)
        breg = bc1[(tid >> 3) * H_DIM + hb + (tid & 7)];

    float c = 0.f;                      // cell state (roles 0,2; threads tid<8)

    for (int s = 0; s < S_STEPS; ++s) {
        const int t = s - role;
        const bool active = (t >= 0) && (t < T_SEQ);
        if (active) {
            // ---- stage h (512 f32) from parity (s+1)&1 of this role's source
            const float* hsrc = (role == 2 ? h2buf : h1buf)
                                + (size_t)((s + 1) & 1) * H_DIM;
            hlds[tid]       = hsrc[tid];
            hlds[tid + 256] = hsrc[tid + 256];
            __syncthreads();

            float sum = 0.f;
#pragma unroll 8
            for (int k = ks * 64; k < ks * 64 + 64; ++k)
                sum += wlds[k][lrow] * hlds[k];    // hlds[k]: wave-wide broadcast
            pt[ks][lrow] = sum;
            __syncthreads();

            if (tid < 32) {
                float g = 0.f;
#pragma unroll
                for (int q = 0; q < 8; ++q) g += pt[q][tid];
                const int grow = (tid >> 3) * H_DIM + hb + (tid & 7);
                if (role == 0) {
                    glds[tid] = g + xg0[(size_t)t * G_DIM + grow];
                    if (t + 8 < T_SEQ)
                        __builtin_prefetch(&xg0[(size_t)(t + 8) * G_DIM + grow], 0, 1);
                } else if (role == 1) {
                    igbuf[(size_t)((s + 1) & 1) * G_DIM + grow] = g;  // parity t&1
                } else {
                    glds[tid] = g + igbuf[(size_t)(s & 1) * G_DIM + grow] + breg;
                }
            }
            __syncthreads();

            if (role != 1 && tid < 8) {
                const float gi = glds[tid],      gf = glds[8 + tid];
                const float gc = glds[16 + tid], go = glds[24 + tid];
                const float i = 1.f / (1.f + __expf(-gi));
                const float f = 1.f / (1.f + __expf(-gf));
                const float g = tanhf(gc);
                const float o = 1.f / (1.f + __expf(-go));
                c = f * c + i * g;
                const float h = o * tanhf(c);
                float* hdst = (role == 0 ? h1buf : h2buf)
                              + (size_t)(s & 1) * H_DIM;     // parity t&1 == s&1
                hdst[hb + tid] = h;
                hnew[tid] = h;
                if (t == T_SEQ - 1) {
                    if (role == 0) { hT0[hb + tid] = h; cT0[hb + tid] = c; }
                    else           { hT1[hb + tid] = h; cT1[hb + tid] = c; }
                }
            }
            __syncthreads();

            if (role == 2 && tid == 0) {
                float s2 = 0.f;
#pragma unroll
                for (int j = 0; j < 8; ++j) s2 += hnew[j] * Wout[hb + j];
                atomicAdd(&outp[t], s2);
            }
        }

        // ---- distributed flag barrier over all 192 WGs ----
        __threadfence();
        __syncthreads();
        if (tid == 0)
            __hip_atomic_store(&flags[wg], s + 1, __ATOMIC_RELEASE,
                               __HIP_MEMORY_SCOPE_AGENT);
        if (tid < 32) {
#pragma unroll
            for (int p = 0; p < 6; ++p) {
                while (__hip_atomic_load(&flags[tid + 32 * p], __ATOMIC_ACQUIRE,
                                         __HIP_MEMORY_SCOPE_AGENT) <= s)
                    __builtin_amdgcn_s_sleep(1);
            }
        }
        __syncthreads();
    }
}

// ---------------------------------------------------------------------------
extern "C" void kernel_launch(void* const* d_in, const int* in_sizes, int n_in,
                              void* d_out, int out_size, void* d_ws, size_t ws_size,
                              hipStream_t stream) {
    const float* in_states = (const float*)d_in[0];
    const float* W_in  = (const float*)d_in[1];
    const float* b_in  = (const float*)d_in[2];
    const float* W_ih0 = (const float*)d_in[3];
    const float* W_hh0 = (const float*)d_in[4];
    const float* b_ih0 = (const float*)d_in[5];
    const float* b_hh0 = (const float*)d_in[6];
    const float* W_ih1 = (const float*)d_in[7];
    const float* W_hh1 = (const float*)d_in[8];
    const float* b_ih1 = (const float*)d_in[9];
    const float* b_hh1 = (const float*)d_in[10];
    const float* W_out = (const float*)d_in[11];
    const float* b_out = (const float*)d_in[12];

    float* out = (float*)d_out;
    float* outputs = out;                  // [8192]
    float* hT0 = out + T_SEQ;              // [512]
    float* hT1 = out + T_SEQ + H_DIM;      // [512]
    float* cT0 = out + T_SEQ + 2 * H_DIM;  // [512]
    float* cT1 = out + T_SEQ + 3 * H_DIM;  // [512]

    // workspace layout (floats)
    float* ws = (float*)d_ws;
    size_t off = 0;
    float* WcT    = ws + off; off += (size_t)G_DIM * IN_DIM;   // [40][2048]
    float* bc0    = ws + off; off += G_DIM;
    float* bc1    = ws + off; off += G_DIM;
    float* h1buf  = ws + off; off += 2 * H_DIM;
    float* h2buf  = ws + off; off += 2 * H_DIM;
    float* igbuf  = ws + off; off += 2 * G_DIM;
    int*   flags  = (int*)(ws + off); off += 256;
    float* inT    = ws + off; off += (size_t)IN_DIM * T_SEQ;   // [40][8192]
    float* xg0    = ws + off; off += (size_t)T_SEQ * G_DIM;    // [8192][2048]
    (void)ws_size; (void)in_sizes; (void)n_in; (void)out_size;

    // 1) fold input projection into layer-0 weights (K-major) + biases
    {
        int total = G_DIM * IN_DIM + 2 * G_DIM;
        prep_kernel<<<(total + 255) / 256, 256, 0, stream>>>(
            W_in, b_in, W_ih0, b_ih0, b_hh0, b_ih1, b_hh1, WcT, bc0, bc1);
    }
    // 2) K-major copy of in_states
    transpose_kernel<<<dim3((IN_DIM + 31) / 32, T_SEQ / 32), 256, 0, stream>>>(
        in_states, inT, T_SEQ, IN_DIM);                        // [8192][40] -> [40][8192]

    // 3) xg0 = in @ Wc^T + bc0   (M=8192, N=2048, K=40) via f32 WMMA
    const int gemmBlocks = ((T_SEQ / 16) * (G_DIM / 64)) / 4;
    wmma_gemm_kmajor<<<gemmBlocks, 128, 0, stream>>>(inT, WcT, bc0, xg0,
                                                     T_SEQ, G_DIM, IN_DIM);

    // 4) fused two-layer pipelined scan (8194 barrier steps instead of 16384)
    reset_kernel<<<8, 256, 0, stream>>>(h1buf, h2buf, flags);
    out_init_kernel<<<T_SEQ / 256, 256, 0, stream>>>(outputs, b_out);
    lstm_fused_scan<<<NWG_ALL, 256, 0, stream>>>(
        xg0, W_hh0, W_ih1, W_hh1, bc1, W_out,
        h1buf, h2buf, igbuf, outputs, hT0, cT0, hT1, cT1, flags);
}